// ConvTransformerBackbone_10213432229986
// MI455X (gfx1250) — compile-verified
//
#include <hip/hip_runtime.h>
#include <hip/hip_bf16.h>

// ---------------- problem constants ----------------
#define BB   2
#define CIN  2048
#define TT   2304
#define EE   512
#define PP_  5800          // padded "channel" length
#define PK   5888          // PP_ rounded up to multiple of 128 (K-tile)
#define TILE_K 128
#define LDS_STRIDE 132     // 16 rows * 132 floats: conflict-free ds_load_b64 pattern
#define SCALE 0.125f       // 64^-0.5

typedef __attribute__((ext_vector_type(2))) float v2f;
typedef __attribute__((ext_vector_type(8))) float v8f;
typedef int i32x4 __attribute__((vector_size(16)));   // matches builtin proto exactly

#define AS1 __attribute__((address_space(1)))
#define AS3 __attribute__((address_space(3)))

// CDNA5 async global->LDS copies (ASYNCcnt-tracked), guarded for portability.
#if defined(__gfx1250__) && __has_builtin(__builtin_amdgcn_global_load_async_to_lds_b128)
#define HAS_ASYNC_LDS 1
#else
#define HAS_ASYNC_LDS 0
#endif

__device__ __forceinline__ void wait_asynccnt0() {
#if __has_builtin(__builtin_amdgcn_s_wait_asynccnt)
    __builtin_amdgcn_s_wait_asynccnt(0);
#else
    asm volatile("s_wait_asynccnt 0x0" ::: "memory");
#endif
}

// ---------------- tiny setup kernels ----------------

// zero the padding tails every launch (ws is poisoned by the harness)
__global__ void init_ws_kernel(float* gbuf, float* qb, float* kb, float* vb, float* hb) {
    int tid = threadIdx.x;
    for (int b = 0; b < BB; ++b) {
        for (int t = TT + tid; t < PK; t += 256) gbuf[(size_t)b * PK + t] = 0.f;
        for (int t = PP_ + tid; t < PK; t += 256) {
            qb[(size_t)b * PK + t] = 0.f;
            kb[(size_t)b * PK + t] = 0.f;
            vb[(size_t)b * PK + t] = 0.f;
            hb[(size_t)b * PK + t] = 0.f;
        }
    }
}

// wbar[c,k] = mean over output channels of w_prj[o,c,k]  (mean commutes with conv)
__global__ void wred_kernel(const float* __restrict__ w_prj, float* __restrict__ wbar) {
    int idx = blockIdx.x * blockDim.x + threadIdx.x;   // c*3+k, 0..6143
    if (idx >= CIN * 3) return;
    float s = 0.f;
    for (int o = 0; o < EE; ++o) s += w_prj[(size_t)o * (CIN * 3) + idx];
    wbar[idx] = s * (1.f / EE);
}

__global__ void bred_kernel(const float* __restrict__ b_prj, float* __restrict__ bbar) {
    __shared__ float sm[256];
    float s = 0.f;
    for (int i = threadIdx.x; i < EE; i += 256) s += b_prj[i];
    sm[threadIdx.x] = s;
    __syncthreads();
    for (int off = 128; off > 0; off >>= 1) {
        if (threadIdx.x < off) sm[threadIdx.x] += sm[threadIdx.x + off];
        __syncthreads();
    }
    if (threadIdx.x == 0) *bbar = sm[0] * (1.f / EE);
}

// g[b,t] = sum_{c,k} x[b,c,t-1+k] * wbar[c,k] + bbar   (fused conv + channel-mean)
__global__ void proj_kernel(const float* __restrict__ x, const float* __restrict__ wbar,
                            const float* __restrict__ bbar, float* __restrict__ gbuf) {
    int t = blockIdx.x * blockDim.x + threadIdx.x;
    int b = blockIdx.y;
    if (t >= TT) return;
    const float* xb = x + (size_t)b * CIN * TT;
    float acc = *bbar;
    for (int c = 0; c < CIN; ++c) {
        float w0 = wbar[c * 3 + 0], w1 = wbar[c * 3 + 1], w2 = wbar[c * 3 + 2];
        const float* xr = xb + (size_t)c * TT;
        float xm = (t > 0)      ? xr[t - 1] : 0.f;
        float x0 = xr[t];
        float xp = (t < TT - 1) ? xr[t + 1] : 0.f;
        acc += xm * w0 + x0 * w1 + xp * w2;
    }
    gbuf[(size_t)b * PK + t] = acc;
}

// ---------------- WMMA batch-2 GEMV: out[b,i] = sum_j W[i,j]*xin[b,j] ----------------
// mode 0: + bias            (q/k/v)
// mode 1: relu(+ bias)      (ff hidden)
// mode 2: + bias + res[b,i] (ff out, residual)
__global__ void matvec_wmma_kernel(const float* __restrict__ W,
                                   const float* __restrict__ xin,
                                   const float* __restrict__ bias,
                                   const float* __restrict__ res,
                                   float* __restrict__ out, int mode) {
    __shared__ float WT[16 * LDS_STRIDE];   // 16 x 128 W tile, padded stride
    __shared__ float XT[BB][TILE_K];        // 2 x 128 xin slice
    __shared__ float red[4][2][16];         // per-wave partials: [wave][batch][row]

    const int tid  = threadIdx.x;
    const int wave = tid >> 5;
    const int lane = tid & 31;
    const int i0   = blockIdx.x * 16;       // 16 output rows per block

    const int n    = lane & 15;             // C/D column index (batch for n<2)
    const int koff = (lane < 16) ? 0 : 2;   // A/B K-lane-half split per ISA layout

    // staging coordinates for the W tile (each thread owns two float4 slots)
    const int r0  = tid >> 5;               // rows 0..7   (slot A)
    const int c40 = tid & 31;
    const int r1  = (tid + 256) >> 5;       // rows 8..15  (slot B)

    v8f acc = {};                           // 16x16 f32 accumulator (cols 0,1 useful)

    for (int j0 = 0; j0 < PP_; j0 += TILE_K) {
        // ---- stage W[i0..i0+15][j0..j0+127] (zero-padded) into LDS ----
        #pragma unroll
        for (int it = 0; it < 2; ++it) {
            int r  = it ? r1 : r0;
            int c4 = c40;
            int gi = i0 + r;
            int gj = j0 + c4 * 4;
            float* dst = &WT[r * LDS_STRIDE + c4 * 4];
            if (gi < PP_ && gj < PP_) {     // PP_ % 4 == 0 -> full-vector guard
#if HAS_ASYNC_LDS
                __builtin_amdgcn_global_load_async_to_lds_b128(
                    (AS1 i32x4*)(W + (size_t)gi * PP_ + gj),
                    (AS3 i32x4*)dst, 0, 0);
#else
                float4 val = *reinterpret_cast<const float4*>(W + (size_t)gi * PP_ + gj);
                dst[0] = val.x; dst[1] = val.y; dst[2] = val.z; dst[3] = val.w;
#endif
            } else {
                dst[0] = 0.f; dst[1] = 0.f; dst[2] = 0.f; dst[3] = 0.f;
            }
        }
        // ---- stage xin[0..1][j0..j0+127] into LDS (tails of xin are zeroed) ----
        {
            int bsel = tid >> 7;            // 0..1
            int c    = tid & 127;
#if HAS_ASYNC_LDS
            __builtin_amdgcn_global_load_async_to_lds_b32(
                (AS1 int*)(xin + (size_t)bsel * PK + j0 + c),
                (AS3 int*)&XT[bsel][c], 0, 0);
#else
            XT[bsel][c] = xin[(size_t)bsel * PK + j0 + c];
#endif
        }
#if HAS_ASYNC_LDS
        wait_asynccnt0();
#endif
        __syncthreads();

        const int cw = wave * 32;           // each wave owns 32 K-columns of the tile
        #pragma unroll
        for (int kk = 0; kk < 8; ++kk) {
            const int cb = cw + kk * 4 + koff;
            v2f a, b;
            // A fragment: row n (both lane halves), K = cb, cb+1
            a.x = WT[n * LDS_STRIDE + cb];
            a.y = WT[n * LDS_STRIDE + cb + 1];
            // B fragment: column n = batch; only n<2 carry data
            float bx = 0.f, by = 0.f;
            if (n < 2) { bx = XT[n][cb]; by = XT[n][cb + 1]; }
            b.x = bx; b.y = by;
            acc = __builtin_amdgcn_wmma_f32_16x16x4_f32(
                      false, a, false, b, (short)0, acc, false, false);
        }
        __syncthreads();
    }

    // lanes 0,1,16,17 hold the useful C columns (batch 0/1), rows r / r+8
    if (n < 2) {
        int mb = (lane < 16) ? 0 : 8;
        #pragma unroll
        for (int r = 0; r < 8; ++r) red[wave][n][mb + r] = acc[r];
    }
    __syncthreads();

    if (tid < 32) {
        int batch = tid >> 4;
        int m     = tid & 15;
        int i     = i0 + m;
        if (i < PP_) {
            float s = red[0][batch][m] + red[1][batch][m] +
                      red[2][batch][m] + red[3][batch][m];
            s += bias[i];
            if (mode == 1)      s = fmaxf(s, 0.f);
            else if (mode == 2) s += res[(size_t)batch * PK + i];
            out[(size_t)batch * PK + i] = s;
        }
    }
}

// ---------------- rank-1 attention ----------------

__global__ void kstats_kernel(const float* __restrict__ kbuf, float* __restrict__ kst) {
    __shared__ float smx[256], smn[256];
    int b = blockIdx.x;
    float mx = -3.4e38f, mn = 3.4e38f;
    for (int j = threadIdx.x; j < PP_; j += 256) {
        float v = kbuf[(size_t)b * PK + j];
        mx = fmaxf(mx, v);
        mn = fminf(mn, v);
    }
    smx[threadIdx.x] = mx; smn[threadIdx.x] = mn;
    __syncthreads();
    for (int off = 128; off > 0; off >>= 1) {
        if (threadIdx.x < off) {
            smx[threadIdx.x] = fmaxf(smx[threadIdx.x], smx[threadIdx.x + off]);
            smn[threadIdx.x] = fminf(smn[threadIdx.x], smn[threadIdx.x + off]);
        }
        __syncthreads();
    }
    if (threadIdx.x == 0) { kst[2 * b] = smx[0]; kst[2 * b + 1] = smn[0]; }
}

// g[b,i] += softmax_j(q_i*scale*k_j) . v   (one block per (i,b) row; in-place residual)
__global__ void attn_kernel(const float* __restrict__ q, const float* __restrict__ k,
                            const float* __restrict__ v, const float* __restrict__ kst,
                            float* __restrict__ g) {
    __shared__ float snum[256], sden[256];
    int i = blockIdx.x;
    int b = blockIdx.y;
    float s = q[(size_t)b * PK + i] * SCALE;
    float m = (s >= 0.f) ? s * kst[2 * b] : s * kst[2 * b + 1];  // row max for stability
    float num = 0.f, den = 0.f;
    for (int j = threadIdx.x; j < PP_; j += 256) {
        float e = __expf(s * k[(size_t)b * PK + j] - m);
        num += e * v[(size_t)b * PK + j];
        den += e;
    }
    snum[threadIdx.x] = num; sden[threadIdx.x] = den;
    __syncthreads();
    for (int off = 128; off > 0; off >>= 1) {
        if (threadIdx.x < off) {
            snum[threadIdx.x] += snum[threadIdx.x + off];
            sden[threadIdx.x] += sden[threadIdx.x + off];
        }
        __syncthreads();
    }
    if (threadIdx.x == 0) g[(size_t)b * PK + i] += snum[0] / sden[0];
}

__global__ void out_kernel(const float* __restrict__ g, float* __restrict__ out) {
    int idx = blockIdx.x * blockDim.x + threadIdx.x;
    if (idx >= BB * TT) return;
    int b = idx / TT, t = idx % TT;
    out[idx] = g[(size_t)b * PK + t];
}

// ---------------- host launcher ----------------
extern "C" void kernel_launch(void* const* d_in, const int* in_sizes, int n_in,
                              void* d_out, int out_size, void* d_ws, size_t ws_size,
                              hipStream_t stream) {
    const float* x      = (const float*)d_in[0];
    // d_in[1] = mask (unused by the global branch)
    const float* w_prj  = (const float*)d_in[2];
    const float* b_prj  = (const float*)d_in[3];
    const float* attn_w = (const float*)d_in[4];
    const float* attn_b = (const float*)d_in[5];
    const float* ff_w   = (const float*)d_in[6];
    const float* ff_b   = (const float*)d_in[7];
    float* out = (float*)d_out;

    float* wsf  = (float*)d_ws;
    float* wbar = wsf;                 // 6144
    float* bbar = wsf + 6144;          // 1
    float* kst  = wsf + 6152;          // 4
    float* gbuf = wsf + 6656;          // 2*PK
    float* qbuf = gbuf + 2 * PK;
    float* kbuf = qbuf + 2 * PK;
    float* vbuf = kbuf + 2 * PK;
    float* hbuf = vbuf + 2 * PK;       // total ~256 KB of ws

    const size_t PPsq = (size_t)PP_ * PP_;
    const int NT = (PP_ + 15) / 16;    // 363 row-tiles

    init_ws_kernel<<<1, 256, 0, stream>>>(gbuf, qbuf, kbuf, vbuf, hbuf);
    wred_kernel<<<(CIN * 3 + 255) / 256, 256, 0, stream>>>(w_prj, wbar);
    bred_kernel<<<1, 256, 0, stream>>>(b_prj, bbar);
    proj_kernel<<<dim3((TT + 255) / 256, BB), 256, 0, stream>>>(x, wbar, bbar, gbuf);

    for (int blk = 0; blk < 2; ++blk) {
        const float* Wq = attn_w + ((size_t)blk * 3 + 0) * PPsq;
        const float* Wk = attn_w + ((size_t)blk * 3 + 1) * PPsq;
        const float* Wv = attn_w + ((size_t)blk * 3 + 2) * PPsq;
        const float* bq = attn_b + ((size_t)blk * 3 + 0) * PP_;
        const float* bk = attn_b + ((size_t)blk * 3 + 1) * PP_;
        const float* bv = attn_b + ((size_t)blk * 3 + 2) * PP_;

        matvec_wmma_kernel<<<NT, 256, 0, stream>>>(Wq, gbuf, bq, nullptr, qbuf, 0);
        matvec_wmma_kernel<<<NT, 256, 0, stream>>>(Wk, gbuf, bk, nullptr, kbuf, 0);
        matvec_wmma_kernel<<<NT, 256, 0, stream>>>(Wv, gbuf, bv, nullptr, vbuf, 0);

        kstats_kernel<<<BB, 256, 0, stream>>>(kbuf, kst);
        attn_kernel<<<dim3(PP_, BB), 256, 0, stream>>>(qbuf, kbuf, vbuf, kst, gbuf);

        const float* W0 = ff_w + ((size_t)blk * 2 + 0) * PPsq;
        const float* W1 = ff_w + ((size_t)blk * 2 + 1) * PPsq;
        const float* f0 = ff_b + ((size_t)blk * 2 + 0) * PP_;
        const float* f1 = ff_b + ((size_t)blk * 2 + 1) * PP_;

        matvec_wmma_kernel<<<NT, 256, 0, stream>>>(W0, gbuf, f0, nullptr, hbuf, 1);
        matvec_wmma_kernel<<<NT, 256, 0, stream>>>(W1, hbuf, f1, gbuf, gbuf, 2);
    }

    out_kernel<<<(BB * TT + 255) / 256, 256, 0, stream>>>(gbuf, out);
}